// ModelNew_10522669875241
// MI455X (gfx1250) — compile-verified
//
#include <hip/hip_runtime.h>
#include <hip/hip_bf16.h>

#define BLOCK 256      // 8 wave32 waves; also rows per block
#define TILE  4096     // fp32 elements per LDS tile (16 KB); x2 buffers = 32 KB LDS

// ---- CDNA5 async global->LDS helpers (gfx1250) --------------------------------
// GLOBAL_LOAD_ASYNC_TO_LDS_B32/B128: per-lane transfer, LDS dest byte-addr in a
// VGPR, 64-bit global addr in a VGPR pair, tracked by ASYNCcnt
// (CDNA5 ISA). One wave B128 instruction moves 512
// contiguous bytes from HBM/L2 straight into LDS with no VGPR staging.
__device__ __forceinline__ unsigned lds_byte_addr(const float* lptr) {
    return (unsigned)(unsigned long long)
           (__attribute__((address_space(3))) const float*)lptr;
}

__device__ __forceinline__ void async_load_b32_to_lds(const float* gptr, float* lptr) {
    asm volatile("global_load_async_to_lds_b32 %0, %1, off"
                 :: "v"(lds_byte_addr(lptr)), "v"((unsigned long long)gptr)
                 : "memory");
}

__device__ __forceinline__ void async_load_b128_to_lds(const float* gptr, float* lptr) {
    asm volatile("global_load_async_to_lds_b128 %0, %1, off"
                 :: "v"(lds_byte_addr(lptr)), "v"((unsigned long long)gptr)
                 : "memory");
}

__device__ __forceinline__ void wait_async0() {
    asm volatile("s_wait_asynccnt 0" ::: "memory");
}

// -------------------------------------------------------------------------------
__global__ __launch_bounds__(BLOCK)
void topk4_csr_kernel(const int* __restrict__ row_ptr,
                      const float* __restrict__ scores,
                      float* __restrict__ out_vals,
                      long long* __restrict__ out_idx,
                      int N) {
    __shared__ __align__(16) float buf[2][TILE];

    const int tid         = threadIdx.x;
    const int firstRow    = blockIdx.x * BLOCK;
    const int lastRowExcl = min(firstRow + BLOCK, N);
    const int blockStart  = row_ptr[firstRow];
    const int blockEnd    = row_ptr[lastRowExcl];
    // 16-byte-align the streaming window (head elems are in-bounds: they belong
    // to earlier rows and are skipped by the per-row clamp below)
    const int alignedStart = blockStart & ~3;

    const int r = firstRow + tid;
    int s = 0, e = 0;
    if (r < N) {
        s = row_ptr[r];
        e = row_ptr[r + 1];
    }

    const float NEG_INF = -__builtin_inff();
    float v0 = NEG_INF, v1 = NEG_INF, v2 = NEG_INF, v3 = NEG_INF;
    int   i0 = -1,      i1 = -1,      i2 = -1,      i3 = -1;

    const int nTiles = (blockEnd - alignedStart + TILE - 1) / TILE;  // uniform per block

    // Fill one LDS tile: bulk in 16B async B128 lane-loads (512 B per wave
    // instruction), ragged tail (<4 elems before blockEnd) in B32 so we never
    // read past row_ptr[N].
    auto issue_tile = [&](int bsel, int base) {
        #pragma unroll
        for (int q = tid; q < TILE / 4; q += BLOCK) {
            const int g = base + q * 4;           // base ≡ 0 (mod 4) -> 16B aligned
            if (g + 4 <= blockEnd) {
                async_load_b128_to_lds(scores + g, &buf[bsel][q * 4]);
            } else if (g < blockEnd) {
                const int rem = blockEnd - g;     // 1..3
                for (int j = 0; j < rem; ++j)
                    async_load_b32_to_lds(scores + g + j, &buf[bsel][q * 4 + j]);
            }
        }
    };

    if (nTiles > 0) {
        // prologue: fill buffer 0 with tile 0
        issue_tile(0, alignedStart);

        for (int t = 0; t < nTiles; ++t) {
            // tile t's async loads must have landed in LDS (each wave waits its
            // own ASYNCcnt; the barrier joins all waves' completions)
            wait_async0();
            __syncthreads();

            // overlap: start tile t+1 into the other buffer while we consume t
            if (t + 1 < nTiles) {
                issue_tile((t + 1) & 1, alignedStart + (t + 1) * TILE);
            }

            // consume tile t: each thread scans its row's slice of this tile
            const int tileBase = alignedStart + t * TILE;
            const int tileEnd  = min(tileBase + TILE, blockEnd);
            const int lo = max(s, tileBase);
            const int hi = min(e, tileEnd);
            const float* lb = buf[t & 1];

            for (int g = lo; g < hi; ++g) {
                float val = lb[g - tileBase];
                // strict '>' insertion => earlier (smaller) index wins ties,
                // exactly matching the reference's tie-breaking
                if (val > v0) {
                    v3 = v2; i3 = i2;
                    v2 = v1; i2 = i1;
                    v1 = v0; i1 = i0;
                    v0 = val; i0 = g;
                } else if (val > v1) {
                    v3 = v2; i3 = i2;
                    v2 = v1; i2 = i1;
                    v1 = val; i1 = g;
                } else if (val > v2) {
                    v3 = v2; i3 = i2;
                    v2 = val; i2 = g;
                } else if (val > v3) {
                    v3 = val; i3 = g;
                }
            }

            // all waves done reading buf[t&1] before it is refilled at t+2
            __syncthreads();
        }
    }

    if (r < N) {
        const long long rr = (long long)r * 4;
        out_vals[rr + 0] = v0;  out_vals[rr + 1] = v1;
        out_vals[rr + 2] = v2;  out_vals[rr + 3] = v3;
        out_idx[rr + 0] = (long long)i0;  out_idx[rr + 1] = (long long)i1;
        out_idx[rr + 2] = (long long)i2;  out_idx[rr + 3] = (long long)i3;
    }
}

// -------------------------------------------------------------------------------
extern "C" void kernel_launch(void* const* d_in, const int* in_sizes, int n_in,
                              void* d_out, int out_size, void* d_ws, size_t ws_size,
                              hipStream_t stream) {
    const int*   row_ptr = (const int*)d_in[0];    // [N+1] int32
    const float* scores  = (const float*)d_in[1];  // [E] float32
    const int N = in_sizes[0] - 1;

    // tuple output, flattened in return order: [N,4] f32 vals, then [N,4] i64 idx
    float*     out_vals = (float*)d_out;
    long long* out_idx  = (long long*)(out_vals + (size_t)N * 4);

    const int grid = (N + BLOCK - 1) / BLOCK;
    hipLaunchKernelGGL(topk4_csr_kernel, dim3(grid), dim3(BLOCK), 0, stream,
                       row_ptr, scores, out_vals, out_idx, N);
}